// LossFunction_16999480557755
// MI455X (gfx1250) — compile-verified
//
#include <hip/hip_runtime.h>
#include <hip/hip_bf16.h>
#include <math.h>

typedef float v2f __attribute__((ext_vector_type(2)));
typedef float v8f __attribute__((ext_vector_type(8)));

#define NB     1024
#define NOUT   512
#define NCLS   5994
#define NCLSP  6000      // padded to multiple of 16
#define NCEN   3
#define NTOPK  5

// margin constants (double-evaluated, written as f32 literals)
#define K_COS_M   0.9800665778412416f    // cos(0.2)
#define K_SIN_M   0.19866933079506122f   // sin(0.2)
#define K_TH      (-0.9800665778412416f) // cos(pi-0.2)
#define K_MM      0.039733866159012244f  // sin(pi-0.2)*0.2
#define K_SCOS_M  0.9982005399352042f    // cos(-0.06)
#define K_SSIN_M  (-0.0599640064794446f) // sin(-0.06)
#define K_STH     (-0.9982005399352042f) // cos(pi+0.06)
#define K_SMM     0.0035978403887666763f // sin(pi+0.06)*(-0.06)
#define K_SCALE   30.0f

__device__ __forceinline__ float phi_f(float c) {
    float s2 = fminf(fmaxf(1.0f - c * c, 0.0f), 1.0f);
    float sine = sqrtf(s2);
    return (c - K_TH > 0.0f) ? (c * K_COS_M - sine * K_SIN_M) : (c - K_MM);
}
__device__ __forceinline__ float sub_phi_f(float c) {
    float s2 = fminf(fmaxf(1.0f - c * c, 0.0f), 1.0f);
    float sine = sqrtf(s2);
    return (c - K_STH > 0.0f) ? (c * K_SCOS_M - sine * K_SSIN_M) : (c - K_SMM);
}

__global__ void init_out_kernel(float* out) {
    out[0] = 0.0f;
    out[1] = 0.0f;
}

// L2-normalize each of the NB rows of x (NOUT=512, 256 thr, 2 elems/thr)
__global__ __launch_bounds__(256) void norm_x_kernel(const float* __restrict__ x,
                                                     float* __restrict__ xn) {
    const int r = blockIdx.x;
    const int t = threadIdx.x;
    const float v0 = x[r * NOUT + t];
    const float v1 = x[r * NOUT + 256 + t];
    __shared__ float s[256];
    s[t] = v0 * v0 + v1 * v1;
    __syncthreads();
    for (int w = 128; w > 0; w >>= 1) {
        if (t < w) s[t] += s[t + w];
        __syncthreads();
    }
    const float inv = 1.0f / fmaxf(sqrtf(s[0]), 1e-12f);
    xn[r * NOUT + t] = v0 * inv;
    xn[r * NOUT + 256 + t] = v1 * inv;
}

// L2-normalize weight rows into padded [NCEN][NCLSP][NOUT]; pad rows -> 0
__global__ __launch_bounds__(256) void norm_w_kernel(const float* __restrict__ w,
                                                     float* __restrict__ wn) {
    const int r = blockIdx.x;            // 0 .. NCEN*NCLSP-1
    const int a = r / NCLSP;
    const int c = r - a * NCLSP;
    const int t = threadIdx.x;
    float* dst = wn + (size_t)r * NOUT;
    if (c >= NCLS) {
        dst[t] = 0.0f;
        dst[256 + t] = 0.0f;
        return;
    }
    const float* src = w + ((size_t)a * NCLS + c) * NOUT;
    const float v0 = src[t];
    const float v1 = src[256 + t];
    __shared__ float s[256];
    s[t] = v0 * v0 + v1 * v1;
    __syncthreads();
    for (int wd = 128; wd > 0; wd >>= 1) {
        if (t < wd) s[t] += s[t + wd];
        __syncthreads();
    }
    const float inv = 1.0f / fmaxf(sqrtf(s[0]), 1e-12f);
    dst[t] = v0 * inv;
    dst[256 + t] = v1 * inv;
}

// One wave per 64(batch) x 16(class) super-tile, x3 subcenters.
// Per 4-wide K step: 7 fragment loads feed 12 v_wmma_f32_16x16x4_f32.
// Fuses max over the 3 subcenters in the epilogue.
__global__ __launch_bounds__(32) void gemm_cos_kernel(const float* __restrict__ xn,
                                                      const float* __restrict__ wn,
                                                      float* __restrict__ cosine) {
    const int ntile = blockIdx.x;   // 0..374  (class tiles, padded N)
    const int mquad = blockIdx.y;   // 0..15   (64 batch rows each)
    const int lane  = threadIdx.x;  // 0..31
    const int mn    = lane & 15;    // A: row M; B: col N
    const int khalf = (lane >> 4) << 1;  // 0 or 2

    const int cls = ntile * 16 + mn;
    const float* A  = xn + ((size_t)(mquad * 64 + mn) * NOUT + khalf);
    const float* B0 = wn + ((size_t)(0 * NCLSP + cls) * NOUT + khalf);
    const float* B1 = wn + ((size_t)(1 * NCLSP + cls) * NOUT + khalf);
    const float* B2 = wn + ((size_t)(2 * NCLSP + cls) * NOUT + khalf);

    v8f acc[4][3];
#pragma unroll
    for (int j = 0; j < 4; ++j)
#pragma unroll
        for (int s = 0; s < 3; ++s)
            acc[j][s] = (v8f){};

#pragma unroll 2
    for (int k = 0; k < NOUT; k += 4) {
        v2f b0 = *(const v2f*)(B0 + k);
        v2f b1 = *(const v2f*)(B1 + k);
        v2f b2 = *(const v2f*)(B2 + k);
#pragma unroll
        for (int j = 0; j < 4; ++j) {
            v2f a = *(const v2f*)(A + (size_t)j * 16 * NOUT + k);
            acc[j][0] = __builtin_amdgcn_wmma_f32_16x16x4_f32(
                false, a, false, b0, (short)0, acc[j][0], false, false);
            acc[j][1] = __builtin_amdgcn_wmma_f32_16x16x4_f32(
                false, a, false, b1, (short)0, acc[j][1], false, false);
            acc[j][2] = __builtin_amdgcn_wmma_f32_16x16x4_f32(
                false, a, false, b2, (short)0, acc[j][2], false, false);
        }
    }

    // C/D layout: VGPR i holds M=i (lanes 0-15) and M=i+8 (lanes 16-31), N=lane&15
    if (cls < NCLS) {
#pragma unroll
        for (int j = 0; j < 4; ++j) {
            const int mbase = mquad * 64 + j * 16 + ((lane >> 4) << 3);
#pragma unroll
            for (int i = 0; i < 8; ++i) {
                float v = fmaxf(fmaxf(acc[j][0][i], acc[j][1][i]), acc[j][2][i]);
                cosine[(size_t)(mbase + i) * NCLS + cls] = v;
            }
        }
    }
}

// Per-row: top-6 cosine, drop label -> top-5 sub set, margin logits,
// streaming log-softmax, accumulate loss & prec1.
__global__ __launch_bounds__(256) void row_loss_kernel(const float* __restrict__ cosine,
                                                       const int* __restrict__ label,
                                                       float* __restrict__ out) {
    const int b = blockIdx.x;
    const int t = threadIdx.x;
    const int lab = label[b];
    const float* row = cosine + (size_t)b * NCLS;

    __shared__ float s_val[256];
    __shared__ int   s_idx[256];
    __shared__ float s_sum[256];

    float topv[6];
    int   topi[6];

#pragma unroll
    for (int k = 0; k < 6; ++k) {
        float bv = -INFINITY;
        int   bi = 0x7fffffff;
        for (int c = t; c < NCLS; c += 256) {
            bool skip = false;
#pragma unroll
            for (int u = 0; u < k; ++u)
                if (topi[u] == c) skip = true;
            float v = row[c];
            if (!skip && (v > bv || (v == bv && c < bi))) { bv = v; bi = c; }
        }
        s_val[t] = bv; s_idx[t] = bi;
        __syncthreads();
        for (int w = 128; w > 0; w >>= 1) {
            if (t < w) {
                float ov = s_val[t + w]; int oi = s_idx[t + w];
                if (ov > s_val[t] || (ov == s_val[t] && oi < s_idx[t])) {
                    s_val[t] = ov; s_idx[t] = oi;
                }
            }
            __syncthreads();
        }
        topv[k] = s_val[0];
        topi[k] = s_idx[0];
        __syncthreads();
    }
    (void)topv;

    // top(k+1) minus label column if present, else drop the last
    int sub[NTOPK];
    int ns = 0;
#pragma unroll
    for (int k = 0; k < 6; ++k) {
        if (topi[k] != lab && ns < NTOPK) sub[ns++] = topi[k];
    }

    // streaming logsumexp of the modified, scaled logits
    float lmax = -INFINITY, lsum = 0.0f;
    for (int c = t; c < NCLS; c += 256) {
        float cs = row[c];
        float v;
        if (c == lab) {
            v = phi_f(cs);
        } else {
            bool insub = false;
#pragma unroll
            for (int u = 0; u < NTOPK; ++u)
                if (sub[u] == c) insub = true;
            v = insub ? sub_phi_f(cs) : cs;
        }
        v *= K_SCALE;
        if (v > lmax) {
            lsum = lsum * __expf(lmax - v) + 1.0f;
            lmax = v;
        } else {
            lsum += __expf(v - lmax);
        }
    }
    s_val[t] = lmax; s_sum[t] = lsum;
    __syncthreads();
    for (int w = 128; w > 0; w >>= 1) {
        if (t < w) {
            float m1 = s_val[t], m2 = s_val[t + w];
            float nm = fmaxf(m1, m2);
            s_sum[t] = s_sum[t] * __expf(m1 - nm) + s_sum[t + w] * __expf(m2 - nm);
            s_val[t] = nm;
        }
        __syncthreads();
    }
    if (t == 0) {
        float lse = s_val[0] + __logf(s_sum[0]);
        float out_lab = K_SCALE * phi_f(row[lab]);
        float loss_b = lse - out_lab;               // -log p[label]
        atomicAdd(&out[0], loss_b * (1.0f / (float)NB));
        float prec = (topi[0] == lab) ? (100.0f / (float)NB) : 0.0f;
        atomicAdd(&out[1], prec);
    }
}

extern "C" void kernel_launch(void* const* d_in, const int* in_sizes, int n_in,
                              void* d_out, int out_size, void* d_ws, size_t ws_size,
                              hipStream_t stream) {
    const float* x = (const float*)d_in[0];      // [1024, 512]
    const float* w = (const float*)d_in[1];      // [3, 5994, 512]
    const int* label = (const int*)d_in[2];      // [1024]
    float* out = (float*)d_out;                  // [loss, prec1]

    float* xn = (float*)d_ws;                               // 1024*512
    float* wn = xn + (size_t)NB * NOUT;                     // 3*6000*512
    float* cosine = wn + (size_t)NCEN * NCLSP * NOUT;       // 1024*5994

    init_out_kernel<<<1, 1, 0, stream>>>(out);
    norm_x_kernel<<<NB, 256, 0, stream>>>(x, xn);
    norm_w_kernel<<<NCEN * NCLSP, 256, 0, stream>>>(w, wn);
    gemm_cos_kernel<<<dim3(NCLSP / 16, NB / 64), 32, 0, stream>>>(xn, wn, cosine);
    row_loss_kernel<<<NB, 256, 0, stream>>>(cosine, label, out);
}